// NearestNeighborSampler_77283641524856
// MI455X (gfx1250) — compile-verified
//
#include <hip/hip_runtime.h>
#include <hip/hip_bf16.h>
#include <limits.h>

typedef __attribute__((ext_vector_type(2))) float v2f;
typedef __attribute__((ext_vector_type(8))) float v8f;

#define BATCH  1024
#define QSZ    32768
#define FDIM   128
#define KNN    5
#define TILE   16
#define WAVES  8
#define NTILES (QSZ / TILE)   // 2048
#define KSTEPS (FDIM / 4)     // 32 wmma_f32_16x16x4 per tile
#define BROW   36             // padded row stride (dwords) in B staging: 16B-aligned, bank-conflict-free
#define CHUNKB 128            // bytes per K=32 chunk within one row
#define TILEB  (WAVES * TILE * FDIM * 4)  // 65536B: global stride between a wave's tiles

// data_queue = concat(data, queue)[:QSZ] -> rows 0..1023 are data, rest queue
__device__ __forceinline__ const float* dq_row(const float* data, const float* queue, int q) {
    return (q < BATCH) ? (data + (size_t)q * FDIM)
                       : (queue + (size_t)(q - BATCH) * FDIM);
}

__device__ __forceinline__ unsigned int lds_off(const void* p) {
    // generic pointers to LDS carry the LDS byte offset in the low 32 bits
    return (unsigned int)(uintptr_t)p;
}

#define WAIT_ASYNC4() asm volatile("s_wait_asynccnt 0x4" ::: "memory")
#define WAIT_ASYNC0() asm volatile("s_wait_asynccnt 0x0" ::: "memory")

__global__ __launch_bounds__(256)
void nns_topk_kernel(const float* __restrict__ data,
                     const float* __restrict__ queue,
                     const int*   __restrict__ nbr,
                     float* __restrict__ out)
{
    // per-wave double-buffered B staging: 2 buffers x 16 rows x 32 floats (padded to BROW)
    __shared__ __align__(16) float bstage[WAVES][2][TILE * BROW];
    __shared__ float simS[WAVES][TILE][TILE + 1];
    __shared__ float mergeV[WAVES][TILE][KNN];
    __shared__ int   mergeI[WAVES][TILE][KNN];
    __shared__ int   sampleIdx[TILE];

    const int tid  = threadIdx.x;
    const int wave = tid >> 5;
    const int lane = tid & 31;
    const int col  = lane & 15;   // N column (or M row for A-frags)
    const int half = lane >> 4;   // K-pair selector within K=4
    const int m0   = blockIdx.x * TILE;

    // ---- preload A fragments (V_WMMA_F32_16X16X4_F32 layout):
    // lane holds A[col][k0], A[col][k0+1] with k0 = 4*s + 2*half
    const float* arow = data + (size_t)(m0 + col) * FDIM + 2 * half;
    v2f   af[KSTEPS];
    float normPart = 0.0f;
#pragma unroll
    for (int s = 0; s < KSTEPS; ++s) {
        af[s] = *(const v2f*)(arow + 4 * s);
        normPart += af[s].x * af[s].x + af[s].y * af[s].y;
    }
    // lanes L and L+16 hold complementary K halves of row `col`
    const float rowNorm = normPart + __shfl_xor(normPart, 16, 32);
    // acc VGPR j corresponds to row m = j + 8*half; that row's norm lives in lane j+8*half
    float sq1[8];
#pragma unroll
    for (int j = 0; j < 8; ++j)
        sq1[j] = __shfl(rowNorm, j + 8 * half, 32);

    // ---- running per-row top-5 (lane m of lanes 0..15 owns row m)
    float thr[8];
#pragma unroll
    for (int j = 0; j < 8; ++j) thr[j] = -INFINITY;
    float lv[KNN]; int li[KNN];
#pragma unroll
    for (int k = 0; k < KNN; ++k) { lv[k] = -INFINITY; li[k] = INT_MAX; }

    const float* buf0 = &bstage[wave][0][0];
    const float* buf1 = &bstage[wave][1][0];
    const int rowA = (lane >> 3);   // + 4*i -> staged row
    const int ccA  = lane & 7;      // 16B chunk within the 128B K-slice of a row

    // fixed LDS destinations (bytes) for the 4 async ops per chunk, per buffer
    unsigned int ldst0[4], ldst1[4];
#pragma unroll
    for (int i = 0; i < 4; ++i) {
        const unsigned int o = (unsigned int)(((4 * i + rowA) * BROW + ccA * 4) * 4);
        ldst0[i] = lds_off(buf0) + o;
        ldst1[i] = lds_off(buf1) + o;
    }

    // persistent per-lane global addresses: addr[i] -> next chunk to prefetch, row group i
    unsigned long long addr[4];
#pragma unroll
    for (int i = 0; i < 4; ++i)
        addr[i] = (unsigned long long)(uintptr_t)
                  (data + (size_t)(wave * TILE + 4 * i + rowA) * FDIM + ccA * 4);

    // issue one chunk (4 x wave-wide 512B async DMA) then advance to the next chunk
    auto issue_chunk = [&](const unsigned int* ldst) {
#pragma unroll
        for (int i = 0; i < 4; ++i) {
            asm volatile("global_load_async_to_lds_b128 %0, %1, off"
                         :: "v"(ldst[i]), "v"(addr[i]) : "memory");
            addr[i] += CHUNKB;
        }
    };

    // prime the pipe: chunks (t=wave, 0) and (t=wave, 1)
    issue_chunk(ldst0);
    issue_chunk(ldst1);

    // ---- scan queue tiles; wave w takes tiles w, w+8, ... (index-ascending per wave)
    for (int t = wave; t < NTILES; t += WAVES) {
        const int  n0      = t * TILE;
        const bool hasNext = (t + WAVES) < NTILES;

        v8f   acc = {};
        float qn  = 0.0f;

        // one K=32 chunk: 8 x (ds_load_b64 + wmma_f32_16x16x4)
        auto compute_chunk = [&](int ci, const float* buf) {
#pragma unroll
            for (int s = 0; s < 8; ++s) {
                v2f b = *(const v2f*)(buf + col * BROW + 4 * s + 2 * half);
                qn += b.x * b.x + b.y * b.y;
                acc = __builtin_amdgcn_wmma_f32_16x16x4_f32(
                          false, af[ci * 8 + s], false, b, (short)0, acc, false, false);
            }
        };

        // chunk stream issued here (in order): (t,2), (t,3), (t+8,0), (t+8,1)
        WAIT_ASYNC4(); compute_chunk(0, buf0); issue_chunk(ldst0);
        WAIT_ASYNC4(); compute_chunk(1, buf1); issue_chunk(ldst1);
        // tile wrap: after (t,3) the stream jumps to (t+8,0): +TILEB from tile start
#pragma unroll
        for (int i = 0; i < 4; ++i) addr[i] += (TILEB - 4 * CHUNKB);
        // data -> queue crossing (wave-uniform, happens exactly once)
        if ((t + WAVES) * TILE == BATCH) {
#pragma unroll
            for (int i = 0; i < 4; ++i)
                addr[i] = (unsigned long long)(uintptr_t)
                          (queue + (size_t)(wave * TILE + 4 * i + rowA) * FDIM + ccA * 4);
        }
        WAIT_ASYNC4(); compute_chunk(2, buf0); if (hasNext) issue_chunk(ldst0);
        if (hasNext) { WAIT_ASYNC4(); } else { WAIT_ASYNC0(); }
        compute_chunk(3, buf1);               if (hasNext) issue_chunk(ldst1);

        const float sq2 = qn + __shfl_xor(qn, 16, 32);  // ||dq_{n0+col}||^2

        float sim[8];
        bool  hit = false;
#pragma unroll
        for (int j = 0; j < 8; ++j) {
            float d2 = sq1[j] + sq2 - 2.0f * acc[j];
            float sv = -sqrtf(fmaxf(d2, 0.0f));
            if (sv == 0.0f) sv = -INFINITY;   // mask self-similarity (== max_similarity)
            sim[j] = sv;
            hit |= (sv > thr[j]);
        }

        if (__any(hit)) {
            // dump tile: row m = j + 8*half, column = col
#pragma unroll
            for (int j = 0; j < 8; ++j)
                simS[wave][j + 8 * half][col] = sim[j];

            float tmine = -INFINITY;
            if (lane < 16) {
                for (int c = 0; c < TILE; ++c) {
                    float v = simS[wave][lane][c];
                    if (v > lv[KNN - 1]) {
                        int id = n0 + c;
#pragma unroll
                        for (int k = 0; k < KNN; ++k) {
                            bool better = (v > lv[k]) || (v == lv[k] && id < li[k]);
                            float tv = lv[k]; int ti = li[k];
                            if (better) { lv[k] = v; li[k] = id; v = tv; id = ti; }
                        }
                    }
                }
                tmine = lv[KNN - 1];
            }
#pragma unroll
            for (int j = 0; j < 8; ++j)
                thr[j] = __shfl(tmine, j + 8 * half, 32);
        }
    }

    // ---- merge the 8 per-wave lists, pick neighbor, gather rows
    if (lane < 16) {
#pragma unroll
        for (int k = 0; k < KNN; ++k) {
            mergeV[wave][lane][k] = lv[k];
            mergeI[wave][lane][k] = li[k];
        }
    }
    __syncthreads();

    if (wave == 0 && lane < 16) {
        float fv[KNN]; int fi[KNN];
#pragma unroll
        for (int k = 0; k < KNN; ++k) { fv[k] = -INFINITY; fi[k] = INT_MAX; }
        for (int ww = 0; ww < WAVES; ++ww) {
#pragma unroll
            for (int k = 0; k < KNN; ++k) {
                float v  = mergeV[ww][lane][k];
                int   id = mergeI[ww][lane][k];
#pragma unroll
                for (int x = 0; x < KNN; ++x) {
                    bool better = (v > fv[x]) || (v == fv[x] && id < fi[x]);
                    float tv = fv[x]; int ti = fi[x];
                    if (better) { fv[x] = v; fi[x] = id; v = tv; id = ti; }
                }
            }
        }
        sampleIdx[lane] = fi[nbr[m0 + lane]];
    }
    __syncthreads();

    // 16 threads per output row, 8 floats each
    {
        const int row = tid >> 4;
        const int c8  = (tid & 15) * 8;
        const float* src = dq_row(data, queue, sampleIdx[row]) + c8;
        float4 a = *(const float4*)(src);
        float4 b = *(const float4*)(src + 4);
        float* dst = out + (size_t)(m0 + row) * FDIM + c8;
        *(float4*)(dst)     = a;
        *(float4*)(dst + 4) = b;
    }
}

extern "C" void kernel_launch(void* const* d_in, const int* in_sizes, int n_in,
                              void* d_out, int out_size, void* d_ws, size_t ws_size,
                              hipStream_t stream) {
    (void)in_sizes; (void)n_in; (void)out_size; (void)d_ws; (void)ws_size;
    const float* data  = (const float*)d_in[0];
    const float* queue = (const float*)d_in[1];
    const int*   nbr   = (const int*)d_in[2];
    float*       out   = (float*)d_out;

    dim3 grid(BATCH / TILE);   // 64 blocks, one 16-row M-tile each
    dim3 block(256);           // 8 wave32s
    nns_topk_kernel<<<grid, block, 0, stream>>>(data, queue, nbr, out);
}